// GVPTProteinFeatures_48000554500382
// MI455X (gfx1250) — compile-verified
//
#include <hip/hip_runtime.h>
#include <hip/hip_bf16.h>
#include <math.h>

typedef float v2f __attribute__((ext_vector_type(2)));
typedef float v8f __attribute__((ext_vector_type(8)));

#define TT 64   // T
#define NN 64   // N
#define LN_EPS 1e-5f

__device__ __forceinline__ v8f wmma_f32_16x16x4(v2f a, v2f b, v8f c) {
    // (neg_a, A, neg_b, B, c_mod, C, reuse_a, reuse_b)
    return __builtin_amdgcn_wmma_f32_16x16x4_f32(false, a, false, b, (short)0, c,
                                                 false, false);
}

__device__ __forceinline__ void norm3(float& x, float& y, float& z) {
    float n = sqrtf(x * x + y * y + z * z);
    float inv = 1.0f / fmaxf(n, 1e-12f);
    x *= inv; y *= inv; z *= inv;
}
__device__ __forceinline__ void cross3(float ax, float ay, float az,
                                       float bx, float by, float bz,
                                       float& cx, float& cy, float& cz) {
    cx = ay * bz - az * by;
    cy = az * bx - ax * bz;
    cz = ax * by - ay * bx;
}

// ---------------------------------------------------------------------------
// Kernel 1: node features + node GVP.  One block per (b,t), 256 threads.
// ---------------------------------------------------------------------------
__global__ __launch_bounds__(256)
void gvp_node_kernel(const float* __restrict__ X,
                     const float* __restrict__ node_wh,    // (3,64)
                     const float* __restrict__ node_ws_w,  // (70,64)
                     const float* __restrict__ node_ws_b,  // (64,)
                     const float* __restrict__ node_wv,    // (64,64)
                     const float* __restrict__ ln_g,
                     const float* __restrict__ ln_b,
                     float* __restrict__ outV) {
    __shared__ float sX[NN * 12];        // X[b,t] : (64 nodes, 4 atoms, 3)
    __shared__ float sU[192 * 3];        // 191 backbone unit vectors
    __shared__ float sDang[192];         // padded dihedral angles
    __shared__ float sV[NN * 9];         // per-node vector triplet v[c][q]
    __shared__ float sAs[NN * 72];       // A: [dih(6) | vn(64) | pad(2)]
    __shared__ float sBwT[64 * 72];      // node_ws_w TRANSPOSED: Bt[col][k], K=72
    __shared__ float sWh[3 * 64];
    __shared__ float sWhv[3 * 64];       // wh @ wv (3,64)
    __shared__ float sSout[NN * 66];     // stride 66
    __shared__ float sMu[NN];
    __shared__ float sInv[NN];

    const int tid = threadIdx.x;
    const int bt  = blockIdx.x;
    const int xbase = bt * NN * 12;

    __builtin_prefetch(node_ws_w, 0, 3);
    __builtin_prefetch(node_wv, 0, 3);

    // ---- Phase 1: loads ----
    for (int i = tid; i < NN * 12; i += 256) sX[i] = X[xbase + i];
    for (int i = tid; i < 192; i += 256)     sWh[i] = node_wh[i];
    for (int i = tid; i < 72 * 64; i += 256) {   // i = k*64 + col
        int k = i >> 6, col = i & 63;
        sBwT[col * 72 + k] = (k < 70) ? node_ws_w[i] : 0.0f;
    }
    __syncthreads();

    // ---- Phase 2: whv = wh @ wv ; backbone unit vectors U ----
    if (tid < 192) {
        int q = tid >> 6, k = tid & 63;
        float acc = 0.0f;
        for (int j = 0; j < 64; ++j) acc += sWh[q * 64 + j] * node_wv[j * 64 + k];
        sWhv[q * 64 + k] = acc;
    }
    if (tid < 191) {  // Xb[m] = atom (m%3) of node (m/3)
        int m = tid;
        int a0 = (m / 3) * 12 + (m % 3) * 3;
        int m1 = m + 1;
        int a1 = (m1 / 3) * 12 + (m1 % 3) * 3;
        float ux = sX[a1 + 0] - sX[a0 + 0];
        float uy = sX[a1 + 1] - sX[a0 + 1];
        float uz = sX[a1 + 2] - sX[a0 + 2];
        norm3(ux, uy, uz);
        sU[m * 3 + 0] = ux; sU[m * 3 + 1] = uy; sU[m * 3 + 2] = uz;
    }
    __syncthreads();

    // ---- Phase 3: dihedral angles ----
    if (tid < 189) {
        const float* u2 = &sU[tid * 3];
        const float* u1 = &sU[(tid + 1) * 3];
        const float* u0 = &sU[(tid + 2) * 3];
        float n2x, n2y, n2z, n1x, n1y, n1z;
        cross3(u2[0], u2[1], u2[2], u1[0], u1[1], u1[2], n2x, n2y, n2z);
        norm3(n2x, n2y, n2z);
        cross3(u1[0], u1[1], u1[2], u0[0], u0[1], u0[2], n1x, n1y, n1z);
        norm3(n1x, n1y, n1z);
        float cosD = n2x * n1x + n2y * n1y + n2z * n1z;
        cosD = fminf(fmaxf(cosD, -1.0f + 1e-7f), 1.0f - 1e-7f);
        float sg = u2[0] * n1x + u2[1] * n1y + u2[2] * n1z;
        float s = (sg > 0.0f) ? 1.0f : ((sg < 0.0f) ? -1.0f : 0.0f);
        sDang[tid + 1] = s * acosf(cosD);
    }
    if (tid == 0) { sDang[0] = 0.0f; sDang[190] = 0.0f; sDang[191] = 0.0f; }
    __syncthreads();

    // ---- Phase 4: per-node vectors + dihedral scalars ----
    if (tid < NN) {
        int n = tid;
        // atoms: 0=N, 1=CA, 2=C
        float cx = sX[n*12 + 6] - sX[n*12 + 3], cy = sX[n*12 + 7] - sX[n*12 + 4], cz = sX[n*12 + 8] - sX[n*12 + 5];
        float nx = sX[n*12 + 0] - sX[n*12 + 3], ny = sX[n*12 + 1] - sX[n*12 + 4], nz = sX[n*12 + 2] - sX[n*12 + 5];
        norm3(cx, cy, cz); norm3(nx, ny, nz);
        float bx = cx + nx, by = cy + ny, bz = cz + nz;
        norm3(bx, by, bz);
        float px, py, pz;
        cross3(cx, cy, cz, nx, ny, nz, px, py, pz);
        norm3(px, py, pz);
        const float w1 = 0.5773502691896258f, w2 = 0.816496580927726f;
        float vx = -bx * w1 - px * w2, vy = -by * w1 - py * w2, vz = -bz * w1 - pz * w2;
        // forward / backward CA directions
        float fx = 0.f, fy = 0.f, fz = 0.f, gx = 0.f, gy = 0.f, gz = 0.f;
        if (n < NN - 1) {
            fx = sX[(n+1)*12+3] - sX[n*12+3];
            fy = sX[(n+1)*12+4] - sX[n*12+4];
            fz = sX[(n+1)*12+5] - sX[n*12+5];
            norm3(fx, fy, fz);
        }
        if (n > 0) {
            gx = sX[n*12+3] - sX[(n-1)*12+3];
            gy = sX[n*12+4] - sX[(n-1)*12+4];
            gz = sX[n*12+5] - sX[(n-1)*12+5];
            norm3(gx, gy, gz);
            gx = -gx; gy = -gy; gz = -gz;
        }
        // v[c][q], q: 0=vec 1=fwd 2=bwd
        sV[n*9 + 0] = vx; sV[n*9 + 1] = fx; sV[n*9 + 2] = gx;
        sV[n*9 + 3] = vy; sV[n*9 + 4] = fy; sV[n*9 + 5] = gy;
        sV[n*9 + 6] = vz; sV[n*9 + 7] = fz; sV[n*9 + 8] = gz;
        // dihedral scalars
        float d0 = sDang[n*3+0], d1 = sDang[n*3+1], d2 = sDang[n*3+2];
        sAs[n*72 + 0] = cosf(d0); sAs[n*72 + 1] = cosf(d1); sAs[n*72 + 2] = cosf(d2);
        sAs[n*72 + 3] = sinf(d0); sAs[n*72 + 4] = sinf(d1); sAs[n*72 + 5] = sinf(d2);
        sAs[n*72 + 70] = 0.0f; sAs[n*72 + 71] = 0.0f;
    }
    __syncthreads();

    // ---- Phase 5: vn[j] = || (v @ wh)[:,j] || ----
    for (int i = tid; i < NN * 64; i += 256) {
        int n = i >> 6, j = i & 63;
        float a0 = 0.f, a1 = 0.f, a2 = 0.f;
        for (int q = 0; q < 3; ++q) {
            float w = sWh[q * 64 + j];
            a0 += sV[n*9 + 0*3 + q] * w;
            a1 += sV[n*9 + 1*3 + q] * w;
            a2 += sV[n*9 + 2*3 + q] * w;
        }
        sAs[n*72 + 6 + j] = sqrtf(a0*a0 + a1*a1 + a2*a2 + 1e-8f);
    }
    __syncthreads();

    // ---- Phase 6: scalar GEMM (64x72)@(72x64) via WMMA f32 16x16x4 ----
    {
        const int wave = tid >> 5, lane = tid & 31;
        const int half = lane >> 4, l15 = lane & 15;
        for (int rep = 0; rep < 2; ++rep) {
            int tile = wave * 2 + rep;          // 0..15
            int mt = tile >> 2, nt = tile & 3;
            v8f acc = {};
            int row = mt * 16 + l15;
            int col = nt * 16 + l15;
            #pragma unroll
            for (int kk = 0; kk < 18; ++kk) {
                int kb = kk * 4 + half * 2;
                v2f a, b;
                a.x = sAs[row * 72 + kb];
                a.y = sAs[row * 72 + kb + 1];
                b.x = sBwT[col * 72 + kb];       // contiguous -> single b64 load
                b.y = sBwT[col * 72 + kb + 1];
                acc = wmma_f32_16x16x4(a, b, acc);
            }
            float bias = node_ws_b[col];
            for (int r = 0; r < 8; ++r) {
                int m = mt * 16 + r + half * 8;
                sSout[m * 66 + col] = acc[r] + bias;
            }
        }
    }
    __syncthreads();

    // ---- Phase 7a: per-node LN statistics ----
    if (tid < NN) {
        int n = tid;
        float mu = 0.f;
        for (int j = 0; j < 64; ++j) mu += sSout[n*66 + j];
        mu *= (1.0f / 64.0f);
        float var = 0.f;
        for (int j = 0; j < 64; ++j) { float d = sSout[n*66 + j] - mu; var += d * d; }
        var *= (1.0f / 64.0f);
        sMu[n]  = mu;
        sInv[n] = rsqrtf(var + LN_EPS);
    }
    __syncthreads();

    // ---- Phase 7b: fully coalesced V output (256 floats per node) ----
    const size_t vbase = (size_t)bt * NN * 256;
    for (int idx = tid; idx < NN * 256; idx += 256) {
        int n = idx >> 8, c = idx & 255;
        float val;
        if (c < 192) {                    // vector part: v @ (wh@wv)
            int cc = c >> 6, k = c & 63;
            float acc = 0.f;
            for (int q = 0; q < 3; ++q) acc += sV[n*9 + cc*3 + q] * sWhv[q * 64 + k];
            val = acc;
        } else {                          // layernormed scalar part
            int j = c - 192;
            val = (sSout[n*66 + j] - sMu[n]) * sInv[n] * ln_g[j] + ln_b[j];
        }
        outV[vbase + idx] = val;
    }
}

// ---------------------------------------------------------------------------
// Kernel 2: per-row neighbor sort + edge features + edge GVP.
// One block per (b,t,i), 64 threads (2 full waves).
// ---------------------------------------------------------------------------
__global__ __launch_bounds__(64)
void gvp_edge_kernel(const float* __restrict__ X,
                     const float* __restrict__ mask,
                     const int* __restrict__ chain_idx,     // (B,L)
                     const int* __restrict__ focuses,       // (B,T,N)
                     const float* __restrict__ edge_wh,     // (1,32)
                     const float* __restrict__ edge_ws_w,   // (64,32)
                     const float* __restrict__ edge_ws_b,   // (32,)
                     const float* __restrict__ edge_wv,     // (32,32)
                     const float* __restrict__ ln_g,
                     const float* __restrict__ ln_b,
                     float* __restrict__ outE,
                     float* __restrict__ outIdx,
                     float* __restrict__ outAbs,
                     int L) {
    __shared__ float sXca[NN * 3];
    __shared__ float sMask[NN];
    __shared__ float sKey[NN];
    __shared__ float sRed[NN];
    __shared__ int   sIdx[NN];
    __shared__ int   sAbs[NN];
    __shared__ int   sChain0;
    __shared__ float sEdir[NN * 3];
    __shared__ float sWh[32];
    __shared__ float sWhv[32];
    __shared__ float sA[NN * 68];     // 64 edges x K=64, stride 68
    __shared__ float sBwT[32 * 68];   // ws_w TRANSPOSED: Bt[col][k], stride 68
    __shared__ float sSout[NN * 36];  // 64 edges x 32, stride 36
    __shared__ float sMu[NN];
    __shared__ float sInv[NN];

    const int tid = threadIdx.x;
    const int id  = blockIdx.x;
    const int i   = id & (NN - 1);    // row within frame
    const int bt  = id >> 6;          // b*T + t
    const int b   = bt / TT;

    __builtin_prefetch(edge_ws_w, 0, 3);
    __builtin_prefetch(edge_wv, 0, 3);

    // ---- loads ----
    {
        int j = tid;
        sXca[j*3 + 0] = X[((size_t)bt * NN + j) * 12 + 3];
        sXca[j*3 + 1] = X[((size_t)bt * NN + j) * 12 + 4];
        sXca[j*3 + 2] = X[((size_t)bt * NN + j) * 12 + 5];
        sMask[j] = mask[(size_t)bt * NN + j];
        if (j < 32) sWh[j] = edge_wh[j];
        for (int k = tid; k < 64 * 32; k += 64) {  // k = kr*32 + col
            int kr = k >> 5, col = k & 31;
            sBwT[col * 68 + kr] = edge_ws_w[k];
        }
    }
    __syncthreads();

    // ---- whv = wh @ wv ; distances ----
    if (tid < 32) {
        float acc = 0.f;
        for (int m = 0; m < 32; ++m) acc += sWh[m] * edge_wv[m * 32 + tid];
        sWhv[tid] = acc;
    }
    const float mi = sMask[i];
    float m2, D;
    {
        int j = tid;
        float dx = sXca[j*3+0] - sXca[i*3+0];
        float dy = sXca[j*3+1] - sXca[i*3+1];
        float dz = sXca[j*3+2] - sXca[i*3+2];
        m2 = mi * sMask[j];
        D  = m2 * sqrtf(dx*dx + dy*dy + dz*dz + 1e-6f);
        sRed[tid] = D;
    }
    __syncthreads();
    for (int s = 32; s > 0; s >>= 1) {
        if (tid < s) sRed[tid] = fmaxf(sRed[tid], sRed[tid + s]);
        __syncthreads();
    }
    {
        float Dmax = sRed[0] + 1.0f;
        sKey[tid] = D + (1.0f - m2) * Dmax;
        sIdx[tid] = tid;
    }

    // ---- bitonic sort (ascending, index tiebreak) ----
    for (int k = 2; k <= NN; k <<= 1) {
        for (int jj = k >> 1; jj > 0; jj >>= 1) {
            __syncthreads();
            int p = tid ^ jj;
            if (p > tid) {
                float k0 = sKey[tid], k1 = sKey[p];
                int   i0 = sIdx[tid], i1 = sIdx[p];
                bool greater = (k0 > k1) || (k0 == k1 && i0 > i1);
                bool asc = ((tid & k) == 0);
                if (greater == asc) {
                    sKey[tid] = k1; sKey[p] = k0;
                    sIdx[tid] = i1; sIdx[p] = i0;
                }
            }
        }
    }
    __syncthreads();

    // ---- gather neighbor metadata ----
    int nb = sIdx[tid];
    int ae = focuses[(size_t)bt * NN + nb];
    sAbs[tid] = ae;
    int ch = chain_idx[(size_t)b * L + ae];
    if (tid == 0) sChain0 = ch;
    __syncthreads();
    float same = (ch == sChain0) ? 1.0f : 0.0f;

    // ---- edge direction + scalar features into A ----
    {
        float dx = sXca[nb*3+0] - sXca[i*3+0];
        float dy = sXca[nb*3+1] - sXca[i*3+1];
        float dz = sXca[nb*3+2] - sXca[i*3+2];
        float n = sqrtf(dx*dx + dy*dy + dz*dz);
        float inv = 1.0f / fmaxf(n, 1e-12f);
        dx *= inv; dy *= inv; dz *= inv;
        sEdir[tid*3+0] = dx; sEdir[tid*3+1] = dy; sEdir[tid*3+2] = dz;
        float vnorm2 = dx*dx + dy*dy + dz*dz;

        float Dn = sKey[tid];
        const float inv_sigma = 1.0f / 1.25f;
        for (int m = 0; m < 16; ++m) {
            float mu = (20.0f / 15.0f) * (float)m;
            float r = (Dn - mu) * inv_sigma;
            sA[tid * 68 + m] = __expf(-r * r);
        }
        float dfl = (float)ae - (float)i;
        const float negln = -9.210340371976184f / 16.0f;  // -ln(10000)/NUM_PE
        for (int p = 0; p < 8; ++p) {
            float freq = __expf((float)(2 * p) * negln);
            float ang = dfl * freq;
            sA[tid * 68 + 16 + p] = __cosf(ang) * same;
            sA[tid * 68 + 24 + p] = __sinf(ang) * same;
        }
        for (int q = 0; q < 32; ++q) {
            float w = sWh[q];
            sA[tid * 68 + 32 + q] = sqrtf(vnorm2 * w * w + 1e-8f);
        }
    }
    __syncthreads();

    // ---- scalar GEMM (64x64)@(64x32) via WMMA f32 16x16x4 ----
    {
        const int wave = tid >> 5, lane = tid & 31;
        const int half = lane >> 4, l15 = lane & 15;
        for (int rep = 0; rep < 4; ++rep) {
            int tile = wave * 4 + rep;     // 0..7
            int mt = tile >> 1, nt = tile & 1;
            v8f acc = {};
            int row = mt * 16 + l15;
            int col = nt * 16 + l15;
            #pragma unroll
            for (int kk = 0; kk < 16; ++kk) {
                int kb = kk * 4 + half * 2;
                v2f a, bb;
                a.x  = sA[row * 68 + kb];
                a.y  = sA[row * 68 + kb + 1];
                bb.x = sBwT[col * 68 + kb];     // contiguous -> single b64 load
                bb.y = sBwT[col * 68 + kb + 1];
                acc = wmma_f32_16x16x4(a, bb, acc);
            }
            float bias = edge_ws_b[col];
            for (int r = 0; r < 8; ++r) {
                int m = mt * 16 + r + half * 8;
                sSout[m * 36 + col] = acc[r] + bias;
            }
        }
    }
    __syncthreads();

    // ---- LN statistics + index outputs ----
    {
        int e = tid;
        float mu = 0.f;
        for (int j = 0; j < 32; ++j) mu += sSout[e*36 + j];
        mu *= (1.0f / 32.0f);
        float var = 0.f;
        for (int j = 0; j < 32; ++j) { float d = sSout[e*36 + j] - mu; var += d * d; }
        var *= (1.0f / 32.0f);
        sMu[e]  = mu;
        sInv[e] = rsqrtf(var + LN_EPS);
        outIdx[(size_t)id * NN + e] = (float)sIdx[e];
        outAbs[(size_t)id * NN + e] = (float)sAbs[e];
    }
    __syncthreads();

    // ---- fully coalesced E output (64 edges x 128 floats) ----
    const size_t ebase = (size_t)id * NN * 128;
    for (int idx = tid; idx < NN * 128; idx += 64) {
        int e = idx >> 7, c = idx & 127;
        float val;
        if (c < 96) {                 // vector part: e_dir x (wh@wv)
            int cc = c >> 5, k = c & 31;
            val = sEdir[e*3 + cc] * sWhv[k];
        } else {                      // layernormed scalar part
            int j = c - 96;
            val = (sSout[e*36 + j] - sMu[e]) * sInv[e] * ln_g[j] + ln_b[j];
        }
        outE[ebase + idx] = val;
    }
}

// ---------------------------------------------------------------------------
extern "C" void kernel_launch(void* const* d_in, const int* in_sizes, int n_in,
                              void* d_out, int out_size, void* d_ws, size_t ws_size,
                              hipStream_t stream) {
    const float* X          = (const float*)d_in[0];
    const float* mask       = (const float*)d_in[1];
    const int*   chain      = (const int*)d_in[2];
    const int*   foc        = (const int*)d_in[3];
    const float* node_wh    = (const float*)d_in[4];
    const float* node_ws_w  = (const float*)d_in[5];
    const float* node_ws_b  = (const float*)d_in[6];
    const float* node_wv    = (const float*)d_in[7];
    const float* edge_wh    = (const float*)d_in[8];
    const float* edge_ws_w  = (const float*)d_in[9];
    const float* edge_ws_b  = (const float*)d_in[10];
    const float* edge_wv    = (const float*)d_in[11];
    const float* ln_n_g     = (const float*)d_in[12];
    const float* ln_n_b     = (const float*)d_in[13];
    const float* ln_e_g     = (const float*)d_in[14];
    const float* ln_e_b     = (const float*)d_in[15];

    const int BT = in_sizes[1] / NN;       // B*T  (mask is B*T*N)
    const int B  = BT / TT;
    const int L  = in_sizes[2] / (B > 0 ? B : 1);

    float* out    = (float*)d_out;
    size_t Velems = (size_t)BT * NN * 256;
    size_t Eelems = (size_t)BT * NN * NN * 128;
    float* outV   = out;
    float* outE   = out + Velems;
    float* outIdx = outE + Eelems;
    float* outAbs = outIdx + (size_t)BT * NN * NN;

    gvp_node_kernel<<<BT, 256, 0, stream>>>(X, node_wh, node_ws_w, node_ws_b,
                                            node_wv, ln_n_g, ln_n_b, outV);
    gvp_edge_kernel<<<BT * NN, 64, 0, stream>>>(X, mask, chain, foc, edge_wh,
                                                edge_ws_w, edge_ws_b, edge_wv,
                                                ln_e_g, ln_e_b,
                                                outE, outIdx, outAbs, L);
}